// GraphAttentionLayer_34660386079361
// MI455X (gfx1250) — compile-verified
//
#include <hip/hip_runtime.h>

// GAT forward for MI455X (gfx1250, wave32).
// GEMMs use V_WMMA_F32_16X16X4_F32 (full fp32 matrix pipe) with weights
// pre-packed into the WMMA B-fragment layout (one float2 load per lane/tile).
// Edge softmax/scatter uses float global atomics (memory-bound phase).

#define IN_DIM  128
#define HEADS   4
#define OUT_DIM 64
#define HD      256   // HEADS*OUT_DIM
#define MLP_HID 256
#define MLP_OUT 128
#define LEAKY   0.2f

typedef __attribute__((ext_vector_type(2))) float v2f;
typedef __attribute__((ext_vector_type(8))) float v8f;

__device__ __forceinline__ v8f wmma4(v2f a, v2f b, v8f c) {
  // V_WMMA_F32_16X16X4_F32: D(16x16 f32) = A(16x4) * B(4x16) + C
  return __builtin_amdgcn_wmma_f32_16x16x4_f32(
      /*neg_a=*/false, a, /*neg_b=*/false, b,
      /*c_mod=*/(short)0, c, /*reuse_a=*/false, /*reuse_b=*/false);
}

// ---------------------------------------------------------------------------
// Packed-B layout: for column-tile t (16 cols), K-step q (4 rows), lane l:
//   Bp[(t*(K/4) + q)*32 + l] = { B[q*4+(l>>4)*2, t*16+(l&15)],
//                                B[q*4+(l>>4)*2+1, t*16+(l&15)] }
// i.e. exactly the per-lane WMMA B fragment, contiguous as float2.
// ---------------------------------------------------------------------------
__global__ void pack_b(const float* __restrict__ B, v2f* __restrict__ Bp,
                       int K, int N) {
  int total = (N >> 4) * (K >> 2) * 32;
  int i = blockIdx.x * blockDim.x + threadIdx.x;
  if (i >= total) return;
  int l  = i & 31;
  int Kq = K >> 2;
  int q  = (i >> 5) % Kq;
  int t  = (i >> 5) / Kq;
  int k  = q * 4 + ((l >> 4) << 1);
  int n  = t * 16 + (l & 15);
  v2f v = { B[(size_t)k * N + n], B[(size_t)(k + 1) * N + n] };
  Bp[i] = v;
}

// Same packing, but source is W_n[H, IN_DIM, OUT_DIM] viewed as
// B[k, n] = W_n[n>>6, k, n&63]  (K=IN_DIM=128, N=HD=256).
__global__ void pack_wn(const float* __restrict__ Wn, v2f* __restrict__ Bp) {
  const int Kq = IN_DIM >> 2;                    // 32
  int total = (HD >> 4) * Kq * 32;               // 16384
  int i = blockIdx.x * blockDim.x + threadIdx.x;
  if (i >= total) return;
  int l = i & 31;
  int q = (i >> 5) % Kq;
  int t = (i >> 5) / Kq;
  int k = q * 4 + ((l >> 4) << 1);
  int n = t * 16 + (l & 15);
  int h = n >> 6, d = n & 63;
  v2f v = { Wn[(size_t)h * IN_DIM * OUT_DIM + (size_t)k * OUT_DIM + d],
            Wn[(size_t)h * IN_DIM * OUT_DIM + (size_t)(k + 1) * OUT_DIM + d] };
  Bp[i] = v;
}

// ---------------------------------------------------------------------------
// GEMM: C[M,N] = A[M,K] @ B[K,N] (+bias, +relu), B pre-packed (see pack_b).
// One wave computes a 16x64 tile (4 WMMA accumulators sharing A loads).
// Requires: M % 16 == 0, N % 64 == 0, K % 4 == 0.
// ---------------------------------------------------------------------------
__global__ void gemm_wmma_f32(const float* __restrict__ A,
                              const v2f* __restrict__ Bp,
                              const float* __restrict__ bias,
                              float* __restrict__ C,
                              int M, int N, int K, int doRelu) {
  const int lane  = threadIdx.x & 31;
  const int wave  = threadIdx.x >> 5;
  const int mtile = blockIdx.x * (blockDim.x >> 5) + wave;
  if (mtile * 16 >= M) return;            // uniform per wave -> EXEC stays all-1s

  const int Kq = K >> 2;
  const int t0 = blockIdx.y * 4;          // first of 4 column tiles (64 cols)
  const int mrow = mtile * 16 + (lane & 15);
  const int ks   = (lane >> 4) << 1;      // lanes 0-15: K+{0,1}; 16-31: K+{2,3}

  v8f acc0 = {}, acc1 = {}, acc2 = {}, acc3 = {};
  const float* Ap = A + (size_t)mrow * K + ks;
  const v2f*   Bq = Bp + (size_t)t0 * Kq * 32 + lane;
  const int ts = Kq * 32;                 // tile stride in v2f elements

#pragma unroll 8
  for (int q = 0; q < Kq; ++q) {
    v2f a  = *(const v2f*)(Ap + 4 * q);   // 8B-aligned (ks even, K%4==0)
    v2f b0 = Bq[0];
    v2f b1 = Bq[ts];
    v2f b2 = Bq[2 * ts];
    v2f b3 = Bq[3 * ts];
    Bq += 32;
    acc0 = wmma4(a, b0, acc0);
    acc1 = wmma4(a, b1, acc1);
    acc2 = wmma4(a, b2, acc2);
    acc3 = wmma4(a, b3, acc3);
  }

  // C/D layout: VGPR r -> row r (lanes 0-15) / row r+8 (lanes 16-31), col = lane&15.
  const int nc    = t0 * 16 + (lane & 15);
  const int rbase = mtile * 16 + ((lane >> 4) << 3);
#pragma unroll
  for (int r = 0; r < 8; ++r) {
    float* Crow = C + (size_t)(rbase + r) * N;
    float v0 = acc0[r], v1 = acc1[r], v2 = acc2[r], v3 = acc3[r];
    if (bias) {
      v0 += bias[nc];      v1 += bias[nc + 16];
      v2 += bias[nc + 32]; v3 += bias[nc + 48];
    }
    if (doRelu) {
      v0 = fmaxf(v0, 0.f); v1 = fmaxf(v1, 0.f);
      v2 = fmaxf(v2, 0.f); v3 = fmaxf(v3, 0.f);
    }
    Crow[nc]      = v0;
    Crow[nc + 16] = v1;
    Crow[nc + 32] = v2;
    Crow[nc + 48] = v3;
  }
}

// Zero num[N*HD], den[N*4], emax keys[N*4].
__global__ void init_buffers(float* __restrict__ num, float* __restrict__ den,
                             unsigned* __restrict__ key, int N) {
  int total = N * HD;
  for (int i = blockIdx.x * blockDim.x + threadIdx.x; i < total;
       i += gridDim.x * blockDim.x) {
    num[i] = 0.f;
    if (i < N * HEADS) { den[i] = 0.f; key[i] = 0u; }
  }
}

// s_src[n,h] = h[n,h,:]·a_src[h,:] ; same for s_dst.
__global__ void node_scores(const float* __restrict__ H,
                            const float* __restrict__ a_src,
                            const float* __restrict__ a_dst,
                            float* __restrict__ ssrc, float* __restrict__ sdst,
                            int N) {
  int t = blockIdx.x * blockDim.x + threadIdx.x;
  if (t >= N * HEADS) return;
  int n = t >> 2, h = t & 3;
  const float* hp = H + (size_t)n * HD + h * OUT_DIM;
  const float* as = a_src + h * OUT_DIM;
  const float* ad = a_dst + h * OUT_DIM;
  float x = 0.f, y = 0.f;
#pragma unroll 8
  for (int d = 0; d < OUT_DIM; ++d) {
    float hv = hp[d];
    x += hv * as[d];
    y += hv * ad[d];
  }
  ssrc[t] = x;
  sdst[t] = y;
}

__device__ __forceinline__ unsigned float_to_ordkey(float f) {
  unsigned u = __float_as_uint(f);
  return (u & 0x80000000u) ? ~u : (u | 0x80000000u);
}
__device__ __forceinline__ float ordkey_to_float(unsigned k) {
  unsigned u = (k & 0x80000000u) ? (k ^ 0x80000000u) : ~k;
  return __uint_as_float(u);
}

// Pass A: segment max of leaky-relu edge scores into orderable uint keys.
__global__ void edge_max(const int* __restrict__ src, const int* __restrict__ dst,
                         const float* __restrict__ ssrc, const float* __restrict__ sdst,
                         unsigned* __restrict__ ekey, int E) {
  int e = blockIdx.x * blockDim.x + threadIdx.x;
  if (e >= E) return;
  int s = src[e], d = dst[e];
#pragma unroll
  for (int h = 0; h < HEADS; ++h) {
    float v = ssrc[s * HEADS + h] + sdst[d * HEADS + h];
    v = (v > 0.f) ? v : LEAKY * v;
    atomicMax(&ekey[d * HEADS + h], float_to_ordkey(v));
  }
}

// Pass B: one wave per (edge, head): ex = exp(e - emax[dst]); scatter-add
// ex*h[src] into num[dst] (64 dims / 32 lanes) and ex into den[dst].
__global__ void edge_scatter(const int* __restrict__ src, const int* __restrict__ dst,
                             const float* __restrict__ ssrc, const float* __restrict__ sdst,
                             const unsigned* __restrict__ ekey,
                             const float* __restrict__ H,
                             float* __restrict__ num, float* __restrict__ den,
                             int E) {
  const int wpb = blockDim.x >> 5;
  long task = (long)blockIdx.x * wpb + (threadIdx.x >> 5);
  if (task >= (long)E * HEADS) return;
  const int lane = threadIdx.x & 31;
  const int e = (int)(task >> 2);
  const int h = (int)(task & 3);
  const int s = src[e], d = dst[e];
  float v = ssrc[s * HEADS + h] + sdst[d * HEADS + h];
  v = (v > 0.f) ? v : LEAKY * v;
  const float emax = ordkey_to_float(ekey[d * HEADS + h]);
  const float ex = expf(v - emax);
  const float* hs = H + (size_t)s * HD + h * OUT_DIM;
  float* nd = num + (size_t)d * HD + h * OUT_DIM;
  atomicAdd(&nd[lane],      ex * hs[lane]);
  atomicAdd(&nd[lane + 32], ex * hs[lane + 32]);
  if (lane == 0) atomicAdd(&den[d * HEADS + h], ex);
}

// In-place: num[n,j] = relu(num[n,j] / safe_den[n, j/64]).
__global__ void activate(float* __restrict__ num, const float* __restrict__ den, int N) {
  int i = blockIdx.x * blockDim.x + threadIdx.x;
  if (i >= N * HD) return;
  int n = i >> 8;
  int h = (i & 255) >> 6;
  float dn = den[n * HEADS + h];
  float sd = (dn > 0.f) ? dn : 1.0f;
  num[i] = fmaxf(num[i] / sd, 0.f);
}

extern "C" void kernel_launch(void* const* d_in, const int* in_sizes, int n_in,
                              void* d_out, int out_size, void* d_ws, size_t ws_size,
                              hipStream_t stream) {
  const float* X    = (const float*)d_in[0];   // [N, 128]
  const int*   src  = (const int*)d_in[1];     // [E]
  const int*   dst  = (const int*)d_in[2];     // [E]
  const float* Wn   = (const float*)d_in[3];   // [4, 128, 64]
  const float* a_s  = (const float*)d_in[4];   // [4, 64]
  const float* a_d  = (const float*)d_in[5];   // [4, 64]
  const float* W1   = (const float*)d_in[6];   // [256, 256]
  const float* b1   = (const float*)d_in[7];   // [256]
  const float* W2   = (const float*)d_in[8];   // [256, 128]
  const float* b2   = (const float*)d_in[9];   // [128]
  float* out = (float*)d_out;                  // [N, 128]

  const int N = in_sizes[0] / IN_DIM;          // 50000
  const int E = in_sizes[1];                   // 800000

  // Workspace layout (floats / v2f share the same arena; all offsets even)
  float* ws    = (float*)d_ws;
  float* Hbuf  = ws;                                  // N*HD  (h; later reused as hidden)
  float* NUM   = Hbuf + (size_t)N * HD;               // N*HD  (num; later act, in place)
  v2f*   WnP   = (v2f*)(NUM + (size_t)N * HD);        // 16384 v2f  (packed W_n)
  v2f*   W1P   = WnP + (size_t)IN_DIM * HD / 2;       // 32768 v2f  (packed W1)
  v2f*   W2P   = W1P + (size_t)HD * MLP_HID / 2;      // 16384 v2f  (packed W2)
  float* Ssrc  = (float*)(W2P + (size_t)MLP_HID * MLP_OUT / 2); // N*4
  float* Sdst  = Ssrc + (size_t)N * HEADS;            // N*4
  unsigned* Ekey = (unsigned*)(Sdst + (size_t)N * HEADS); // N*4
  float* Den   = (float*)(Ekey + (size_t)N * HEADS);  // N*4

  const int mtiles = (N + 15) / 16;                   // 3125
  const int WPB = 4;                                  // waves (m-tiles) per block
  dim3 gblock(32 * WPB);

  // 1) pack weights into WMMA B-fragment order
  pack_wn<<<(IN_DIM * HD / 2 + 255) / 256, 256, 0, stream>>>(Wn, WnP);
  pack_b<<<(HD * MLP_HID / 2 + 255) / 256, 256, 0, stream>>>(W1, W1P, HD, MLP_HID);
  pack_b<<<(MLP_HID * MLP_OUT / 2 + 255) / 256, 256, 0, stream>>>(W2, W2P, MLP_HID, MLP_OUT);

  // 2) zero accumulators / keys
  init_buffers<<<4096, 256, 0, stream>>>(NUM, Den, Ekey, N);

  // 3) h = X @ W   [N,128]x[128,256]
  {
    dim3 g((mtiles + WPB - 1) / WPB, HD / 64);
    gemm_wmma_f32<<<g, gblock, 0, stream>>>(X, WnP, nullptr, Hbuf, N, HD, IN_DIM, 0);
  }

  // 4) node attention scores
  node_scores<<<(N * HEADS + 255) / 256, 256, 0, stream>>>(Hbuf, a_s, a_d, Ssrc, Sdst, N);

  // 5) segment max over incoming edges
  edge_max<<<(E + 255) / 256, 256, 0, stream>>>(src, dst, Ssrc, Sdst, Ekey, E);

  // 6) softmax numerator/denominator scatter
  {
    long tasks = (long)E * HEADS;            // 3.2M waves
    int blocks = (int)((tasks + 7) / 8);     // 8 waves / 256-thread block
    edge_scatter<<<blocks, 256, 0, stream>>>(src, dst, Ssrc, Sdst, Ekey, Hbuf, NUM, Den, E);
  }

  // 7) head_out = relu(num / safe_den), in place -> NUM is now the concat features
  activate<<<(N * HD + 255) / 256, 256, 0, stream>>>(NUM, Den, N);

  // 8) hidden = relu(concat @ W1 + b1)   (hidden reuses Hbuf)
  {
    dim3 g((mtiles + WPB - 1) / WPB, MLP_HID / 64);
    gemm_wmma_f32<<<g, gblock, 0, stream>>>(NUM, W1P, b1, Hbuf, N, MLP_HID, HD, 1);
  }

  // 9) out = hidden @ W2 + b2
  {
    dim3 g((mtiles + WPB - 1) / WPB, MLP_OUT / 64);
    gemm_wmma_f32<<<g, gblock, 0, stream>>>(Hbuf, W2P, b2, out, N, MLP_OUT, MLP_HID, 0);
  }
}